// MAELoss_9543417331714
// MI455X (gfx1250) — compile-verified
//
#include <hip/hip_runtime.h>

typedef __attribute__((ext_vector_type(4))) float v4f;
typedef __attribute__((ext_vector_type(2))) float v2f;
typedef __attribute__((ext_vector_type(8))) float v8f;

// log1p thresholds from the reference
#define THR1 1.7917594692280550f   // log1p(5)
#define THR2 3.2580965380214821f   // log1p(25)
#define THR3 3.9318256327243257f   // log1p(50)

__device__ __forceinline__ float weight_of(float y) {
    return (y < THR1) ? 0.2f
         : (y < THR2) ? 30.0f
         : (y < THR3) ? 2500.0f
                      : 20000.0f;
}

// Wave32 all-lane sum via V_WMMA_F32_16X16X4_F32.
// A (16x4 f32): lane L, VGPR0 holds A[M=L%16][K = L<16 ? 0 : 2]; VGPR1 -> K=1/3.
// We place v in VGPR0, 0 in VGPR1. B = all-ones (layout-immune).
// D[i][j] = sum_k A[i][k] = v_lane_i + v_lane_{i+16}, replicated over j.
// D layout: VGPR r = row r (lanes 0-15) / row r+8 (lanes 16-31), so
// sum(d[0..7]) gives rows 0-7 total on lanes 0-15 and rows 8-15 total on
// lanes 16-31; a SWAPX16 ds_swizzle fold yields the full 32-lane sum.
// Must be called with all 32 lanes of the wave active (EXEC all-1s).
__device__ __forceinline__ float wave_reduce_wmma(float v) {
    v2f a; a.x = v;    a.y = 0.0f;
    v2f b; b.x = 1.0f; b.y = 1.0f;
    v8f c = {};
    v8f d = __builtin_amdgcn_wmma_f32_16x16x4_f32(
        /*neg_a=*/false, a, /*neg_b=*/false, b,
        /*c_mod=*/(short)0, c, /*reuse_a=*/false, /*reuse_b=*/false);
    float s = d[0] + d[1] + d[2] + d[3] + d[4] + d[5] + d[6] + d[7];
    // SWAPX16: xor=0x10, or=0x00, and=0x1f  ->  offset 0x401F
    int other = __builtin_amdgcn_ds_swizzle(__float_as_int(s), 0x401F);
    return s + __int_as_float(other);
}

// Reduce (num, den) over a 256-thread block. Tree in LDS down to one wave,
// then WMMA wave reduction.
__device__ __forceinline__ float2 block_reduce_pair(float num, float den) {
    __shared__ float snum[256];
    __shared__ float sden[256];
    __shared__ float res[2];
    const int tid = threadIdx.x;
    snum[tid] = num; sden[tid] = den;
    __syncthreads();
    if (tid < 128) { snum[tid] += snum[tid + 128]; sden[tid] += sden[tid + 128]; }
    __syncthreads();
    if (tid < 64)  { snum[tid] += snum[tid + 64];  sden[tid] += sden[tid + 64];  }
    __syncthreads();
    if (tid < 32) {  // wave 0, all 32 lanes active -> EXEC all-1s for WMMA
        float n2 = snum[tid] + snum[tid + 32];
        float d2 = sden[tid] + sden[tid + 32];
        float nt = wave_reduce_wmma(n2);
        float dt = wave_reduce_wmma(d2);
        if (tid == 0) { res[0] = nt; res[1] = dt; }
    }
    __syncthreads();
    return make_float2(res[0], res[1]);
}

// Pass 1: streaming weighted-MAE partial sums. Bandwidth-bound:
// float4 non-temporal loads, grid-stride, fp32 accumulation.
__global__ void __launch_bounds__(256)
wmae_partial_kernel(const float* __restrict__ y_pred,
                    const float* __restrict__ y_true,
                    float* __restrict__ wnum, float* __restrict__ wden,
                    long n4, long n) {
    const long tid    = (long)blockIdx.x * blockDim.x + threadIdx.x;
    const long stride = (long)gridDim.x * blockDim.x;
    const v4f* p4 = (const v4f*)y_pred;
    const v4f* t4 = (const v4f*)y_true;

    float num = 0.0f, den = 0.0f;
    for (long i = tid; i < n4; i += stride) {
        v4f t = __builtin_nontemporal_load(&t4[i]);
        v4f p = __builtin_nontemporal_load(&p4[i]);
#pragma unroll
        for (int k = 0; k < 4; ++k) {
            float w = weight_of(t[k]);
            num = fmaf(w, fabsf(t[k] - p[k]), num);
            den += w;
        }
    }
    // scalar tail (n not divisible by 4)
    for (long i = n4 * 4 + tid; i < n; i += stride) {
        float t = y_true[i], p = y_pred[i];
        float w = weight_of(t);
        num = fmaf(w, fabsf(t - p), num);
        den += w;
    }

    float2 r = block_reduce_pair(num, den);
    if (threadIdx.x == 0) {
        wnum[blockIdx.x] = r.x;
        wden[blockIdx.x] = r.y;
    }
}

// Pass 2: reduce block partials, write out = num/den.
__global__ void __launch_bounds__(256)
wmae_final_kernel(const float* __restrict__ wnum,
                  const float* __restrict__ wden,
                  int nb, float* __restrict__ out) {
    float num = 0.0f, den = 0.0f;
    for (int i = threadIdx.x; i < nb; i += blockDim.x) {
        num += wnum[i];
        den += wden[i];
    }
    float2 r = block_reduce_pair(num, den);
    if (threadIdx.x == 0) out[0] = r.x / r.y;
}

extern "C" void kernel_launch(void* const* d_in, const int* in_sizes, int n_in,
                              void* d_out, int out_size, void* d_ws, size_t ws_size,
                              hipStream_t stream) {
    const float* y_pred = (const float*)d_in[0];
    const float* y_true = (const float*)d_in[1];
    float* out = (float*)d_out;

    const long n  = (long)in_sizes[0];
    const long n4 = n / 4;

    // partials workspace: wnum[NB] followed by wden[NB]
    int NB = 2048;
    const long cap = (long)(ws_size / (2 * sizeof(float)));
    if ((long)NB > cap) NB = (int)(cap > 0 ? cap : 1);

    float* wnum = (float*)d_ws;
    float* wden = wnum + NB;

    wmae_partial_kernel<<<NB, 256, 0, stream>>>(y_pred, y_true, wnum, wden, n4, n);
    wmae_final_kernel<<<1, 256, 0, stream>>>(wnum, wden, NB, out);
}